// SphericalHarmonicEdgeAttrs_spin_soc_simple_87213605913075
// MI455X (gfx1250) — compile-verified
//
#include <hip/hip_runtime.h>
#include <cstdint>

// Edges per block == blockDim (8 wave32 waves).
#define EPB 256
#define TILE_FLOATS (EPB * 27)          // 6912 floats = 27648 B, 16B-aligned tiles
#define TILE_BYTES  (TILE_FLOATS * 4)
#define NQ          (TILE_FLOATS / 4)   // 1728 16-byte DMA chunks per tile

// Real spherical harmonics l=1 (3) and l=2 (5), e3nn 'component' normalization.
__device__ __forceinline__ void sh_l12(float x, float y, float z,
                                       float* __restrict__ o1,
                                       float* __restrict__ o2) {
  constexpr float S3  = 1.7320508075688772f;   // sqrt(3)
  constexpr float S5  = 2.2360679774997896f;   // sqrt(5)
  constexpr float S15 = 3.8729833462074170f;   // sqrt(15)
  o1[0] = S3 * x;
  o1[1] = S3 * y;
  o1[2] = S3 * z;
  o2[0] = S15 * x * z;
  o2[1] = S15 * x * y;
  o2[2] = S5 * (y * y - 0.5f * (x * x + z * z));
  o2[3] = S15 * y * z;
  o2[4] = 0.5f * S15 * (z * z - x * x);
}

__global__ __launch_bounds__(EPB) void edge_sh27_kernel(
    const float* __restrict__ pos,    // [N,3]
    const float* __restrict__ spin,   // [N,3]
    const int*   __restrict__ eidx,   // [2,E] (row = eidx[e], col = eidx[E+e])
    float*       __restrict__ out,    // [E,27] in SORT_LIST order
    int E) {
  // Double-buffered staging tile: async DMA drains buf ^1 while we fill buf.
  __shared__ float tile[2][TILE_FLOATS];   // 54 KB of the WGP's 320 KB LDS
  const int tid = threadIdx.x;
  // Raw LDS byte offset of the tiles (low 32 bits of the flat shared pointer).
  const uint32_t lds0 = (uint32_t)(uintptr_t)&tile[0][0];

  const int numTiles = (E + EPB - 1) / EPB;
  int it = 0;
  for (int t = blockIdx.x; t < numTiles; t += gridDim.x, ++it) {
    const int buf = it & 1;
    const uint32_t lbase = lds0 + (uint32_t)buf * TILE_BYTES;
    const int e0 = t * EPB;
    const int e  = e0 + tid;
    const bool active = (e < E);

    // Prefetch the index stream for this block's next tile
    // (global_prefetch_b8: fire-and-forget, no counter).
    {
      const long pn = (long)e0 + (long)gridDim.x * EPB + tid;
      if (pn < (long)E) {
        __builtin_prefetch(&eidx[pn], 0, 0);
        __builtin_prefetch(&eidx[(long)E + pn], 0, 0);
      }
    }

    // ---- Issue all input loads BEFORE draining the old DMA: the L2 gather
    // latency overlaps with the in-flight async stores of iteration it-1.
    int r = 0, c = 0;
    float pix = 0, piy = 0, piz = 0, pjx = 0, pjy = 0, pjz = 0;
    float six = 0, siy = 0, siz = 0, sjx = 0, sjy = 0, sjz = 0;
    if (active) {
      r = eidx[e];        // row (edge source)
      c = eidx[E + e];    // col (edge target)
      pix = pos[3 * r + 0]; piy = pos[3 * r + 1]; piz = pos[3 * r + 2];
      pjx = pos[3 * c + 0]; pjy = pos[3 * c + 1]; pjz = pos[3 * c + 2];
      six = spin[3 * c + 0]; siy = spin[3 * c + 1]; siz = spin[3 * c + 2];
      sjx = spin[3 * r + 0]; sjy = spin[3 * r + 1]; sjz = spin[3 * r + 2];
    }

    // ---- Retire async stores that read tile[buf] (issued at iteration it-2).
    // Each full tile issues 6-7 chunks/thread, so ASYNCcnt <= 6 guarantees only
    // the previous iteration's stores (other buffer) can still be in flight.
    asm volatile("s_wait_asynccnt 6" ::: "memory");
    __syncthreads();   // extend that guarantee across all waves in the group

    if (active) {
      float vx = pjx - pix, vy = pjy - piy, vz = pjz - piz;
      const float inv = 1.0f / sqrtf(vx * vx + vy * vy + vz * vz);
      vx *= inv; vy *= inv; vz *= inv;

      float r1[3], r2[5], a1[3], a2[5], b1[3], b2[5];
      sh_l12(vx, vy, vz, r1, r2);          // rij (normalized)
      sh_l12(six, siy, siz, a1, a2);       // si = spin[col] (unnormalized)
      sh_l12(sjx, sjy, sjz, b1, b2);       // sj = spin[row] (unnormalized)

      // 27 outputs in sorted-irrep order, row-major into the LDS tile.
      // Stride-27 rows: bank = (tid*27 + k) % 64, gcd(27,64)=1 -> conflict-free.
      float* o = &tile[buf][tid * 27];
      o[0] = 1.0f; o[1] = 1.0f; o[2] = 1.0f;              // l=0: rij, si, sj
      o[3] = r1[0];  o[4] = r1[1];  o[5] = r1[2];         // l=1 rij
      o[6] = a1[0];  o[7] = a1[1];  o[8] = a1[2];         // l=1 si
      o[9] = b1[0];  o[10] = b1[1]; o[11] = b1[2];        // l=1 sj
#pragma unroll
      for (int k = 0; k < 5; ++k) {                       // l=2 rij | si | sj
        o[12 + k] = r2[k];
        o[17 + k] = a2[k];
        o[22 + k] = b2[k];
      }
    }
    __syncthreads();   // tile[buf] fully written

    const int valid = (E - e0) < EPB ? (E - e0) : EPB;
    float* gbase = out + (size_t)e0 * 27;

    if (valid == EPB) {
      // Full tile: stream LDS -> global via the CDNA5 async DMA path.
      // Tile base byte offset = t * 27648 -> 16B aligned, pure b128 chunks.
      // Deliberately NOT waited here: drained at iteration it+2 (same buffer).
      const uint64_t g = (uint64_t)(uintptr_t)gbase;
#pragma unroll 4
      for (int i = tid; i < NQ; i += EPB) {
        asm volatile("global_store_async_from_lds_b128 %0, %1, off"
                     :
                     : "v"(g + (uint64_t)i * 16u),
                       "v"(lbase + (uint32_t)i * 16u)
                     : "memory");
      }
    } else {
      // Tail tile (E % 256 != 0): plain coalesced b32 stores (synchronous).
      for (int i = tid; i < valid * 27; i += EPB) gbase[i] = tile[buf][i];
    }
    // No trailing barrier needed: reuse of tile[buf] is gated by the
    // s_wait_asynccnt + barrier at the top of iteration it+2.
  }
  // Outstanding async stores are drained by S_ENDPGM's implicit wait-idle.
}

extern "C" void kernel_launch(void* const* d_in, const int* in_sizes, int n_in,
                              void* d_out, int out_size, void* d_ws, size_t ws_size,
                              hipStream_t stream) {
  (void)n_in; (void)out_size; (void)d_ws; (void)ws_size;
  const float* pos  = (const float*)d_in[0];   // [N,3] f32
  const float* spin = (const float*)d_in[1];   // [N,3] f32
  const int*   eidx = (const int*)d_in[2];     // [2,E] int32 (harness convention)
  float*       out  = (float*)d_out;           // [E,27] f32

  const int E = in_sizes[2] / 2;
  const int numTiles = (E + EPB - 1) / EPB;
  const int grid = numTiles < 2048 ? numTiles : 2048;  // persistent blocks

  edge_sh27_kernel<<<grid, EPB, 0, stream>>>(pos, spin, eidx, out, E);
}